// GCNConv_gqcml_88794153877678
// MI455X (gfx1250) — compile-verified
//
#include <hip/hip_runtime.h>

typedef __attribute__((ext_vector_type(2))) float v2f;
typedef __attribute__((ext_vector_type(4))) float v4f;
typedef __attribute__((ext_vector_type(8))) float v8f;

#define NNODES 512
#define BLOCK  256

// One block per batch. 8 waves x 4 row-tiles x 16 rows = 512 rows.
// agg = adj(512x512) @ X(512x2) via V_WMMA_F32_16X16X4_F32.
// K-slot permutation trick: slots {0,1,2,3} of WMMA #0 take K {k,k+1,k+4,k+5},
// WMMA #1 takes {k+2,k+3,k+6,k+7}. This makes both the A fragment (global) and
// the B fragment (LDS) a single contiguous b128 load per lane per 2 WMMAs.
// Lanes with N>=2 read a zeroed LDS region (B columns 2..15 are zero padding),
// so there is no per-step masking VALU in the hot loop.
__global__ __launch_bounds__(BLOCK) void gcn_fused_kernel(
    const float* __restrict__ X,      // [B,512,2]
    const float* __restrict__ Adj,    // [B,512,512]
    const float* __restrict__ W1,     // [2,4]
    const float* __restrict__ B1,     // [4]
    const float* __restrict__ W2,     // [4,1]
    const float* __restrict__ B2,     // [1]
    float* __restrict__ out)          // [B,1]
{
    __shared__ float xt[2 * NNODES];   // transposed features: xt[f*512 + k]
    __shared__ float zbuf[NNODES];     // zeros for B columns 2..15
    __shared__ float red[8];

    const int b    = blockIdx.x;
    const int tid  = threadIdx.x;
    const int lane = tid & 31;
    const int wave = tid >> 5;
    const int n    = lane & 15;     // N column of the 16-wide tile (0,1 valid)
    const int half = lane >> 4;     // 0: K slots 0,1 ; 1: K slots 2,3

    // Stage X[b] into LDS transposed (coalesced reads), and zero the pad region.
    const float* Xb = X + (size_t)b * (NNODES * 2);
    for (int i = tid; i < NNODES * 2; i += BLOCK) {
        xt[(i & 1) * NNODES + (i >> 1)] = Xb[i];
    }
    for (int i = tid; i < NNODES; i += BLOCK) {
        zbuf[i] = 0.0f;
    }
    __syncthreads();

    // Tiny MLP weights in registers.
    float w1a[4], w1b[4], b1v[4], w2v[4];
    #pragma unroll
    for (int j = 0; j < 4; ++j) {
        w1a[j] = W1[j];        // W1[0][j]
        w1b[j] = W1[4 + j];    // W1[1][j]
        b1v[j] = B1[j];
        w2v[j] = W2[j];
    }
    const float b2s = B2[0];

    const float* adjB = Adj + (size_t)b * NNODES * NNODES;

    // B-matrix LDS source: column 0/1 of X for lanes n<2, zeros otherwise.
    const float* bsrc = (n < 2) ? &xt[n * NNODES + 4 * half] : &zbuf[4 * half];

    float waveSum = 0.0f;

    for (int t = wave; t < NNODES / 16; t += 8) {
        const int row = t * 16 + n;
        const float* asrc = adjB + (size_t)row * NNODES + 4 * half;

        v8f c0 = {};
        v8f c1 = {};

        // 128 WMMAs per tile; one b128 global + one b128 LDS load per 2 WMMAs.
        #pragma unroll 4
        for (int k = 0; k < NNODES; k += 8) {
            v4f a4 = __builtin_nontemporal_load((const v4f*)(asrc + k)); // streamed once
            v4f b4 = *(const v4f*)(bsrc + k);
            v2f alo = __builtin_shufflevector(a4, a4, 0, 1);
            v2f ahi = __builtin_shufflevector(a4, a4, 2, 3);
            v2f blo = __builtin_shufflevector(b4, b4, 0, 1);
            v2f bhi = __builtin_shufflevector(b4, b4, 2, 3);
            c0 = __builtin_amdgcn_wmma_f32_16x16x4_f32(false, alo, false, blo,
                                                       (short)0, c0, false, false);
            c1 = __builtin_amdgcn_wmma_f32_16x16x4_f32(false, ahi, false, bhi,
                                                       (short)0, c1, false, false);
        }

        // C/D layout: VGPR r, lanes 0-15 -> M=r, lanes 16-31 -> M=r+8; lane = N.
        float partial = 0.0f;
        #pragma unroll
        for (int r = 0; r < 8; ++r) {
            float cc = c0[r] + c1[r];
            float a0 = cc;                       // agg[:,0] on lanes with N==0
            float a1 = __shfl_xor(cc, 1, 32);    // agg[:,1] pulled from N==1 lane
            float p = 0.0f;
            #pragma unroll
            for (int j = 0; j < 4; ++j) {
                float h = fmaf(a0, w1a[j], fmaf(a1, w1b[j], b1v[j]));
                h = fmaxf(h, 0.0f);              // ReLU
                p = fmaf(h, w2v[j], p);          // h @ W2
            }
            partial += p;
        }
        if (n != 0) partial = 0.0f;              // keep only lanes 0 and 16
        partial += __shfl_xor(partial, 16, 32);  // rows 0-7 + rows 8-15
        waveSum += partial;
    }

    if (lane == 0) red[wave] = waveSum;
    __syncthreads();
    if (tid == 0) {
        float tot = 0.0f;
        #pragma unroll
        for (int w = 0; w < 8; ++w) tot += red[w];
        out[b] = tot + (float)NNODES * b2s;      // + N * b2 from the readout bias
    }
}

extern "C" void kernel_launch(void* const* d_in, const int* in_sizes, int n_in,
                              void* d_out, int out_size, void* d_ws, size_t ws_size,
                              hipStream_t stream) {
    (void)in_sizes; (void)n_in; (void)d_ws; (void)ws_size;
    const float* X  = (const float*)d_in[0];
    const float* A  = (const float*)d_in[1];
    const float* W1 = (const float*)d_in[2];
    const float* B1 = (const float*)d_in[3];
    const float* W2 = (const float*)d_in[4];
    const float* B2 = (const float*)d_in[5];
    float* out = (float*)d_out;
    const int B = out_size;                      // 256 batches, out = [B,1]
    gcn_fused_kernel<<<B, BLOCK, 0, stream>>>(X, A, W1, B1, W2, B2, out);
}